// GraphSAGE_20444044329487
// MI455X (gfx1250) — compile-verified
//
#include <hip/hip_runtime.h>

typedef __attribute__((ext_vector_type(2))) float v2f;
typedef __attribute__((ext_vector_type(8))) float v8f;

#define F 64            // feature width (both layers)
#define KDIM 128        // 2*F GEMM reduction dim
#define NKT (KDIM / 4)  // 32 k-tiles of depth 4

// ---------------------------------------------------------------- zero scratch
__global__ __launch_bounds__(256) void zero_f32(float* __restrict__ p, size_t n) {
    size_t i = (size_t)blockIdx.x * blockDim.x + threadIdx.x;
    size_t stride = (size_t)gridDim.x * blockDim.x;
    for (; i < n; i += stride) p[i] = 0.0f;
}

// ---------------------------------------------------------------- degree count
__global__ __launch_bounds__(256) void count_edges(const int* __restrict__ edst,
                                                   float* __restrict__ cnt, int E) {
    int i = blockIdx.x * blockDim.x + threadIdx.x;
    int stride = gridDim.x * blockDim.x;
    for (; i < E; i += stride) atomicAdd(&cnt[edst[i]], 1.0f);
}

// ---------------------------------------------------------------- edge scatter-sum
// One wave32 per edge: lanes cover features [lane] and [32+lane].
// Gather is coalesced (128B per half-row); sum uses L2-side f32 atomics.
__global__ __launch_bounds__(256) void scatter_sum(const float* __restrict__ x,
                                                   const int* __restrict__ esrc,
                                                   const int* __restrict__ edst,
                                                   float* __restrict__ sum, int E) {
    int wave   = (blockIdx.x * blockDim.x + threadIdx.x) >> 5;
    int lane   = threadIdx.x & 31;
    int nwaves = (gridDim.x * blockDim.x) >> 5;
    for (int e = wave; e < E; e += nwaves) {
        int s = esrc[e], d = edst[e];
        const float* xs = x + (size_t)s * F;
        float v0 = xs[lane];
        float v1 = xs[32 + lane];
        float* sd = sum + (size_t)d * F;
        atomicAdd(sd + lane, v0);
        atomicAdd(sd + 32 + lane, v1);
    }
}

// ---------------------------------------------------------------- sum -> mean (in place)
__global__ __launch_bounds__(256) void mean_div(float* __restrict__ sum,
                                                const float* __restrict__ cnt, size_t n64) {
    size_t i = (size_t)blockIdx.x * blockDim.x + threadIdx.x;
    size_t stride = (size_t)gridDim.x * blockDim.x;
    for (; i < n64; i += stride) {
        float c = cnt[i >> 6];
        sum[i] = sum[i] / (c > 1.0f ? c : 1.0f);
    }
}

// ---------------------------------------------------------------- SAGE layer GEMM
// out[m, n] = relu( sum_k concat(self,mean)[m,k] * W[k,n] + b[n] )
// One wave per 16-row tile; 4 accumulators cover all 64 output columns.
// V_WMMA_F32_16X16X4_F32 (full f32 precision). W is staged once per block into
// LDS in fragment-packed order, so each B fragment is a single ds_load_b64.
__global__ __launch_bounds__(256) void sage_gemm(const float* __restrict__ self,  // [N,64]
                                                 const float* __restrict__ mean,  // [N,64]
                                                 const float* __restrict__ W,     // [128,64]
                                                 const float* __restrict__ b,     // [64]
                                                 float* __restrict__ out,         // [N,64]
                                                 int n_tiles) {
    __shared__ float wlds[KDIM * F];  // 32 KB: packed B fragments

    int lane = threadIdx.x & 31;
    int half = lane >> 4;                 // 0: lanes 0-15, 1: lanes 16-31
    int lr   = lane & 15;

    // Stage W: entry p = (t*4+j)*32 + l holds float2 {W[(4t+2h)*64+col], W[(4t+2h+1)*64+col]}
    // where h = l>>4, col = j*16 + (l&15). All waves participate before any exits.
    for (int p = threadIdx.x; p < NKT * 4 * 32; p += 256) {
        int l = p & 31;
        int j = (p >> 5) & 3;
        int t = p >> 7;
        int col = j * 16 + (l & 15);
        int krow = 4 * t + 2 * (l >> 4);
        wlds[2 * p]     = W[(size_t)krow * F + col];
        wlds[2 * p + 1] = W[(size_t)(krow + 1) * F + col];
    }
    __syncthreads();

    int wave = (blockIdx.x * blockDim.x + threadIdx.x) >> 5;
    if (wave >= n_tiles) return;          // uniform per-wave exit; EXEC stays all-1s
    int row0 = wave * 16;

    // Bias into accumulators: C layout -> every vgpr of acc[j] holds col j*16+lr.
    v8f acc[4];
#pragma unroll
    for (int j = 0; j < 4; ++j) {
        float bj = b[j * 16 + lr];
#pragma unroll
        for (int i = 0; i < 8; ++i) acc[j][i] = bj;
    }

    // A fragment: lane lr row M=row0+lr, K = {k0+2h, k0+2h+1} -> one global float2 load.
    // B fragment: one ds_load_b64 at immediate offset (t*4+j)*256 + lane*8.
    auto kstep = [&](const float* __restrict__ src, int ksrc, int t) {
        v2f a = *(const v2f*)(src + (size_t)(row0 + lr) * F + ksrc + 2 * half);
#pragma unroll
        for (int j = 0; j < 4; ++j) {
            v2f bb = *(const v2f*)(&wlds[((t * 4 + j) * 32 + lane) * 2]);
            acc[j] = __builtin_amdgcn_wmma_f32_16x16x4_f32(
                false, a, false, bb, (short)0, acc[j], false, false);
        }
    };

#pragma unroll 4
    for (int k0 = 0; k0 < F; k0 += 4) kstep(self, k0, k0 >> 2);        // W rows 0..63
#pragma unroll 4
    for (int k0 = 0; k0 < F; k0 += 4) kstep(mean, k0, (k0 >> 2) + 16); // W rows 64..127

    // D layout: vgpr i -> row row0+i (lo lanes) / row0+i+8 (hi lanes), col j*16+lr.
#pragma unroll
    for (int j = 0; j < 4; ++j)
#pragma unroll
        for (int i = 0; i < 8; ++i) {
            float v = acc[j][i];
            out[(size_t)(row0 + i + 8 * half) * F + j * 16 + lr] = v > 0.0f ? v : 0.0f;
        }
}

// ---------------------------------------------------------------- launch
extern "C" void kernel_launch(void* const* d_in, const int* in_sizes, int n_in,
                              void* d_out, int out_size, void* d_ws, size_t ws_size,
                              hipStream_t stream) {
    const float* x    = (const float*)d_in[0];
    const int*   esrc = (const int*)  d_in[1];
    const int*   edst = (const int*)  d_in[2];
    const float* W1   = (const float*)d_in[3];
    const float* b1   = (const float*)d_in[4];
    const float* W2   = (const float*)d_in[5];
    const float* b2   = (const float*)d_in[6];
    float* out = (float*)d_out;

    const int N = in_sizes[0] / F;     // 100000
    const int E = in_sizes[1];         // 1600000

    float* sum = (float*)d_ws;         // [N*64]
    float* cnt = sum + (size_t)N * F;  // [N]

    const size_t nSum = (size_t)N * F;
    const int n_tiles = N / 16;                  // N is a multiple of 16
    const int gemm_blocks = (n_tiles + 7) / 8;   // 8 waves per 256-thread block

    // ---- degree counts (shared by both layers) + layer-1 aggregation
    zero_f32   <<<2048, 256, 0, stream>>>(sum, nSum + N);          // sum and cnt contiguous
    count_edges<<<2048, 256, 0, stream>>>(edst, cnt, E);
    scatter_sum<<<4096, 256, 0, stream>>>(x, esrc, edst, sum, E);
    mean_div   <<<2048, 256, 0, stream>>>(sum, cnt, nSum);
    sage_gemm  <<<gemm_blocks, 256, 0, stream>>>(x, sum, W1, b1, out, n_tiles);

    // ---- layer 2 (input = out, in-place safe: self rows only read by their own wave)
    zero_f32   <<<2048, 256, 0, stream>>>(sum, nSum);
    scatter_sum<<<4096, 256, 0, stream>>>(out, esrc, edst, sum, E);
    mean_div   <<<2048, 256, 0, stream>>>(sum, cnt, nSum);
    sage_gemm  <<<gemm_blocks, 256, 0, stream>>>(out, sum, W2, b2, out, n_tiles);
}